// ApproxCompressor_57062935495406
// MI455X (gfx1250) — compile-verified
//
#include <hip/hip_runtime.h>
#include <math.h>

// ---------------- problem constants ----------------
#define NB        32          // batches
#define CH        2           // channels
#define LSIG      131072      // samples per channel
#define SEG_LEN   4096        // samples per segment (one block)
#define SEGS      (LSIG / SEG_LEN)       // 32 segments per batch
#define NSEG_TOT  (NB * SEGS)            // 1024 blocks
#define TPB       256
#define WAVES     (TPB / 32)             // 8 waves / block (wave32)
#define WSPAN     (SEG_LEN / WAVES)      // 512 samples per wave
#define STEPS     (WSPAN / 256)          // 2 WMMA macro-steps per wave

typedef float v2f __attribute__((ext_vector_type(2)));
typedef float v8f __attribute__((ext_vector_type(8)));

__device__ __forceinline__ float softplusf(float x) {
    if (x > 20.f)  return x;
    if (x < -20.f) return expf(x);
    return log1pf(expf(x));
}

__device__ __forceinline__ float4 mul4(float4 g, float4 a) {
    return make_float4(g.x * a.x, g.y * a.y, g.z * a.z, g.w * a.w);
}

// =====================================================================
// Kernel 1: per-segment decay-weighted tail sum (zero-IC local end value)
//   seg_end[n,s] = (1-a) * sum_j a^(SEG_LEN-1-j) * loud[n, s*SEG_LEN + j]
// =====================================================================
__global__ __launch_bounds__(TPB) void k_seg_tail(
    const float* __restrict__ x, const float* __restrict__ z_alpha,
    float* __restrict__ seg_end)
{
    const int blk = blockIdx.x;
    const int n = blk / SEGS, s = blk % SEGS;
    const int tid = threadIdx.x, lane = tid & 31, w = tid >> 5;

    const float alpha = 1.f / (1.f + expf(-z_alpha[n]));
    const float lg2a  = log2f(alpha);
    const float onem  = 1.f - alpha;

    const float4* x0 = (const float4*)(x + ((size_t)n * CH + 0) * LSIG + (size_t)s * SEG_LEN);
    const float4* x1 = (const float4*)(x + ((size_t)n * CH + 1) * LSIG + (size_t)s * SEG_LEN);
    __builtin_prefetch(x0, 0, 3);   // global_prefetch_b8: warm L2 for the streaming read
    __builtin_prefetch(x1, 0, 3);

    float wsum = 0.f;
    for (int i4 = tid; i4 < SEG_LEN / 4; i4 += TPB) {
        float4 a = x0[i4], b = x1[i4];
        const int bi = i4 * 4;
        float l0 = 0.5f * (a.x * a.x + b.x * b.x);
        float l1 = 0.5f * (a.y * a.y + b.y * b.y);
        float l2 = 0.5f * (a.z * a.z + b.z * b.z);
        float l3 = 0.5f * (a.w * a.w + b.w * b.w);
        wsum += l0 * exp2f((float)(SEG_LEN - 1 - (bi + 0)) * lg2a);
        wsum += l1 * exp2f((float)(SEG_LEN - 1 - (bi + 1)) * lg2a);
        wsum += l2 * exp2f((float)(SEG_LEN - 1 - (bi + 2)) * lg2a);
        wsum += l3 * exp2f((float)(SEG_LEN - 1 - (bi + 3)) * lg2a);
    }
    // wave reduce, then block reduce
    for (int o = 16; o > 0; o >>= 1) wsum += __shfl_down(wsum, o, 32);
    __shared__ float red[WAVES];
    if (lane == 0) red[w] = wsum;
    __syncthreads();
    if (tid == 0) {
        float t = 0.f;
        for (int i = 0; i < WAVES; ++i) t += red[i];
        seg_end[n * SEGS + s] = onem * t;
    }
}

// =====================================================================
// Kernel 2: scan of segment carries per batch (32 lanes, one per batch)
//   carry_in[n,s] = env value just before segment s starts
// =====================================================================
__global__ void k_carry_scan(
    const float* __restrict__ seg_end, const float* __restrict__ z_alpha,
    float* __restrict__ carry_in)
{
    const int n = threadIdx.x;
    if (n >= NB) return;
    const float alpha = 1.f / (1.f + expf(-z_alpha[n]));
    const float aSeg  = exp2f((float)SEG_LEN * log2f(alpha));   // alpha^4096
    float c = 0.f;
    for (int s = 0; s < SEGS; ++s) {
        carry_in[n * SEGS + s] = c;
        c = aSeg * c + seg_end[n * SEGS + s];
    }
}

// =====================================================================
// Kernel 3: main pass.  Per segment: LDS-stage input, wave-level carry
// hierarchy, WMMA (f32 16x16x4) blocked scan of 256-sample tiles,
// gain computation, coalesced b128 output.
// =====================================================================
__global__ __launch_bounds__(TPB) void k_compress(
    const float* __restrict__ x,
    const float* __restrict__ z_alpha, const float* __restrict__ log_T,
    const float* __restrict__ log_ratio, const float* __restrict__ log_knee,
    const float* __restrict__ carry_in, float* __restrict__ out)
{
    __shared__ __align__(16) float x0s[SEG_LEN];
    __shared__ __align__(16) float x1s[SEG_LEN];
    __shared__ __align__(16) float gbuf[WAVES * 256];
    __shared__ float wave_end[WAVES];
    __shared__ float wave_carry[WAVES];

    const int blk = blockIdx.x;
    const int n = blk / SEGS, s = blk % SEGS;
    const int tid = threadIdx.x, lane = tid & 31, w = tid >> 5;
    const int half = lane >> 4, cidx = lane & 15;

    // per-batch parameters
    const float alpha = 1.f / (1.f + expf(-z_alpha[n]));
    const float lg2a  = log2f(alpha);
    const float onem  = 1.f - alpha;
    const float Tv    = log_T[n] - 6.f;
    const float ratio = 1.f + expf(log_ratio[n]);
    const float kneeV = expf(log_knee[n] - 1.f);
    const float coefA = (1.f / ratio - 1.f) / kneeV;

    // -------- phase A: stage input segment into LDS (coalesced b128) --------
    const float* x0p = x + ((size_t)n * CH + 0) * LSIG + (size_t)s * SEG_LEN;
    const float* x1p = x + ((size_t)n * CH + 1) * LSIG + (size_t)s * SEG_LEN;
    {
        float4* d0 = (float4*)x0s;
        float4* d1 = (float4*)x1s;
        const float4* s0 = (const float4*)x0p;
        const float4* s1 = (const float4*)x1p;
        for (int i4 = tid; i4 < SEG_LEN / 4; i4 += TPB) { d0[i4] = s0[i4]; d1[i4] = s1[i4]; }
    }
    __syncthreads();

    // per-wave decay-weighted tail over its 512-sample sub-segment
    float wsum = 0.f;
    for (int j = lane; j < WSPAN; j += 32) {
        const int idx = w * WSPAN + j;
        float p = x0s[idx], q = x1s[idx];
        wsum += 0.5f * (p * p + q * q) * exp2f((float)(WSPAN - 1 - j) * lg2a);
    }
    for (int o = 16; o > 0; o >>= 1) wsum += __shfl_down(wsum, o, 32);
    if (lane == 0) wave_end[w] = onem * wsum;
    __syncthreads();

    // -------- phase B: scan the 8 wave carries --------
    if (tid == 0) {
        const float a512 = exp2f((float)WSPAN * lg2a);
        float c = carry_in[n * SEGS + s];
        for (int q = 0; q < WAVES; ++q) { wave_carry[q] = c; c = a512 * c + wave_end[q]; }
    }
    __syncthreads();

    // -------- phase C: WMMA blocked scan --------
    // A[m][j] = (j<=m) ? (1-a)*a^(m-j) : 0   (16x16, split into 4 K-slices of 4)
    // ISA A layout (16x4 f32): lanes 0-15 & 16-31 both map M=lane&15;
    // VGPR comp 0/1 hold K = 2*half + {0,1} of the slice.
    v2f aslice[4];
    #pragma unroll
    for (int kk = 0; kk < 4; ++kk) {
        const int j0 = 4 * kk + 2 * half;
        float ax = (j0     <= cidx) ? onem * exp2f((float)(cidx - j0)     * lg2a) : 0.f;
        float ay = (j0 + 1 <= cidx) ? onem * exp2f((float)(cidx - j0 - 1) * lg2a) : 0.f;
        v2f av; av.x = ax; av.y = ay; aslice[kk] = av;
    }
    float pw[8];
    #pragma unroll
    for (int r = 0; r < 8; ++r) pw[r] = exp2f((float)(r + 8 * half + 1) * lg2a);
    const float a16 = exp2f(16.f * lg2a);

    float carry = wave_carry[w];
    for (int step = 0; step < STEPS; ++step) {
        const int off = w * WSPAN + step * 256;

        // D[i][c] = local env (zero-IC) of sample i of chunk c; 4 chained WMMAs over K
        v8f acc = {0.f, 0.f, 0.f, 0.f, 0.f, 0.f, 0.f, 0.f};
        #pragma unroll
        for (int kk = 0; kk < 4; ++kk) {
            const int i0 = off + cidx * 16 + 4 * kk + 2 * half;
            float p0 = x0s[i0],     q0 = x1s[i0];
            float p1 = x0s[i0 + 1], q1 = x1s[i0 + 1];
            v2f bv; bv.x = 0.5f * (p0 * p0 + q0 * q0);
                    bv.y = 0.5f * (p1 * p1 + q1 * q1);
            acc = __builtin_amdgcn_wmma_f32_16x16x4_f32(
                false, aslice[kk], false, bv, (short)0, acc, false, false);
        }

        // chunk ends: row M=15 lives in VGPR7 of lanes 16..31
        float ce = __shfl(acc[7], 16 + cidx, 32);
        // (a,b) linear-recurrence scan over 16 chunks: x_end = a16*x_in + ce
        float sa = a16, sb = ce;
        #pragma unroll
        for (int d = 1; d < 16; d <<= 1) {
            float pa = __shfl_up(sa, (unsigned)d, 16);
            float pb = __shfl_up(sb, (unsigned)d, 16);
            if (cidx >= d) { sb = sa * pb + sb; sa = sa * pa; }
        }
        const float incl = sa * carry + sb;              // env at end of chunk cidx
        const float prevIncl = __shfl_up(incl, 1u, 16);
        const float chunk_carry = (cidx == 0) ? carry : prevIncl;
        const float carry_next  = __shfl(incl, 15, 32);

        // env -> gain, staged to LDS for coalesced write-out
        #pragma unroll
        for (int r = 0; r < 8; ++r) {
            float env = acc[r] + pw[r] * chunk_carry;
            float le  = logf(env + 1e-5f);
            float g   = expf(coefA * softplusf(kneeV * (le - Tv)));
            gbuf[w * 256 + cidx * 16 + (r + 8 * half)] = g;
        }
        carry = carry_next;
        __syncthreads();

        // coalesced b128 output: lane handles 8 contiguous samples
        {
            const float4* gv  = (const float4*)(gbuf + w * 256 + lane * 8);
            const float4* a0v = (const float4*)(x0s + off + lane * 8);
            const float4* a1v = (const float4*)(x1s + off + lane * 8);
            float4 g0 = gv[0],  g1 = gv[1];
            float4 c00 = a0v[0], c01 = a0v[1];
            float4 c10 = a1v[0], c11 = a1v[1];
            const size_t ob = (size_t)s * SEG_LEN + off + lane * 8;
            float4* o0 = (float4*)(out + ((size_t)n * CH + 0) * LSIG + ob);
            float4* o1 = (float4*)(out + ((size_t)n * CH + 1) * LSIG + ob);
            o0[0] = mul4(g0, c00); o0[1] = mul4(g1, c01);
            o1[0] = mul4(g0, c10); o1[1] = mul4(g1, c11);
        }
        __syncthreads();
    }
}

// =====================================================================
extern "C" void kernel_launch(void* const* d_in, const int* in_sizes, int n_in,
                              void* d_out, int out_size, void* d_ws, size_t ws_size,
                              hipStream_t stream)
{
    const float* x  = (const float*)d_in[0];
    const float* za = (const float*)d_in[1];
    const float* lt = (const float*)d_in[2];
    const float* lr = (const float*)d_in[3];
    const float* lk = (const float*)d_in[4];
    float* out = (float*)d_out;

    float* seg_end = (float*)d_ws;           // NSEG_TOT floats
    float* carry   = seg_end + NSEG_TOT;     // NSEG_TOT floats  (8 KB total)

    hipLaunchKernelGGL(k_seg_tail,   dim3(NSEG_TOT), dim3(TPB), 0, stream, x, za, seg_end);
    hipLaunchKernelGGL(k_carry_scan, dim3(1),        dim3(32),  0, stream, seg_end, za, carry);
    hipLaunchKernelGGL(k_compress,   dim3(NSEG_TOT), dim3(TPB), 0, stream,
                       x, za, lt, lr, lk, carry, out);
}